// SpatialAttention_MCU_65618510348370
// MI455X (gfx1250) — compile-verified
//
#include <hip/hip_runtime.h>
#include <hip/hip_bf16.h>

typedef __attribute__((ext_vector_type(2))) float v2f;
typedef __attribute__((ext_vector_type(4))) float v4f;
typedef __attribute__((ext_vector_type(8))) float v8f;

#define B_  16
#define C_  256
#define H_  128
#define W_  128
#define HW_ (H_ * W_)          // 16384
#define CHW_ (C_ * HW_)        // 4194304
#define NPOS_ (B_ * HW_)       // 262144
#define NTOT_ (B_ * CHW_)      // 67108864

// ---------------------------------------------------------------------------
// Pass 1: channel mean (via V_WMMA_F32_16X16X4_F32 with all-ones A) + channel
// max (VALU on the same loads). One wave -> 16 consecutive spatial positions.
// B-matrix block = 4 channels x 16 positions per WMMA; A = ones so every row
// of D is the 4-channel column sum; accumulate 64 WMMAs over C=256.
// ---------------------------------------------------------------------------
__global__ void sa_reduce_kernel(const float* __restrict__ x,
                                 float* __restrict__ avg_out,
                                 float* __restrict__ max_out) {
    const int lane  = threadIdx.x & 31;
    const int wave  = threadIdx.x >> 5;
    const int group = blockIdx.x * 8 + wave;      // 16 positions per group
    const int pos0  = group << 4;
    const int b     = pos0 >> 14;                 // / HW_
    const int hw0   = pos0 & (HW_ - 1);
    const float* xb = x + ((size_t)b * CHW_) + hw0;

    const int half = lane >> 4;                   // 0 | 1
    const int col  = lane & 15;                   // N index (position)

    v2f a; a.x = 1.0f; a.y = 1.0f;                // ones A-matrix (16x4)
    v8f acc = {};
    float mx = -3.4028235e38f;

    #pragma unroll 4
    for (int c = 0; c < C_; c += 4) {
        // this lane contributes channels {c+half, c+2+half} for position col
        float b0 = xb[(size_t)(c + half)     * HW_ + col];
        float b1 = xb[(size_t)(c + 2 + half) * HW_ + col];
        v2f bb; bb.x = b0; bb.y = b1;
        acc = __builtin_amdgcn_wmma_f32_16x16x4_f32(
            /*neg_a=*/false, a, /*neg_b=*/false, bb,
            /*c_mod=*/(short)0, acc, /*reuse_a=*/false, /*reuse_b=*/false);
        mx = fmaxf(mx, fmaxf(b0, b1));
    }
    // halves saw channels {0,2} vs {1,3} mod 4 -> combine across halves
    mx = fmaxf(mx, __shfl_xor(mx, 16, 32));

    // D layout: VGPR0, lanes 0-15 hold M=0, N=lane (all rows equal here)
    if (lane < 16) {
        avg_out[pos0 + col] = acc[0] * (1.0f / (float)C_);
        max_out[pos0 + col] = mx;
    }
}

// ---------------------------------------------------------------------------
// Pass 2: 3x3 conv over [avg, max] (zero-padded) + hardsigmoid -> y map.
// Tiny (2 MiB read / 1 MiB write), weights are uniform -> SMEM-cached.
// ---------------------------------------------------------------------------
__global__ void sa_conv_kernel(const float* __restrict__ avgm,
                               const float* __restrict__ maxm,
                               const float* __restrict__ wgt,   // [1][2][3][3]
                               float* __restrict__ y) {
    const int p  = blockIdx.x * blockDim.x + threadIdx.x;       // 0..NPOS_-1
    const int b  = p >> 14;
    const int hw = p & (HW_ - 1);
    const int h  = hw >> 7;
    const int w  = hw & (W_ - 1);
    const float* a = avgm + ((size_t)b << 14);
    const float* m = maxm + ((size_t)b << 14);

    float acc = 0.0f;
    #pragma unroll
    for (int dh = -1; dh <= 1; ++dh) {
        const int hh = h + dh;
        if (hh < 0 || hh >= H_) continue;
        #pragma unroll
        for (int dw = -1; dw <= 1; ++dw) {
            const int ww = w + dw;
            if (ww < 0 || ww >= W_) continue;
            const int q  = (hh << 7) + ww;
            const int wi = (dh + 1) * 3 + (dw + 1);
            acc += wgt[wi] * a[q] + wgt[9 + wi] * m[q];
        }
    }
    // hardsigmoid = clip(v+3, 0, 6) / 6
    y[p] = fminf(fmaxf(acc + 3.0f, 0.0f), 6.0f) * (1.0f / 6.0f);
}

// ---------------------------------------------------------------------------
// Pass 3: out = x * y, float4 vectorized. Reverse block order so the tail of
// x still resident in the 192 MB L2 (from pass 1's stream) is reused before
// eviction; NT loads for x (last use) and NT stores for out (never re-read).
// y (1 MiB) stays L2-resident with default temporal policy.
// ---------------------------------------------------------------------------
__global__ void sa_scale_kernel(const float* __restrict__ x,
                                const float* __restrict__ y,
                                float* __restrict__ out) {
    // reversed block order, ascending within block (keeps full coalescing)
    const size_t i4   = (size_t)(gridDim.x - 1 - blockIdx.x) * blockDim.x
                        + threadIdx.x;
    const size_t base = i4 * 4;                  // flat element index
    const size_t b    = base >> 22;              // / CHW_
    const size_t hw   = base & (HW_ - 1);        // 4 consecutive w, same b,c

    v4f xv = __builtin_nontemporal_load((const v4f*)(x + base));
    v4f yv = *(const v4f*)(y + (b << 14) + hw);
    __builtin_nontemporal_store(xv * yv, (v4f*)(out + base));
}

extern "C" void kernel_launch(void* const* d_in, const int* in_sizes, int n_in,
                              void* d_out, int out_size, void* d_ws, size_t ws_size,
                              hipStream_t stream) {
    (void)in_sizes; (void)n_in; (void)out_size; (void)ws_size;
    const float* x      = (const float*)d_in[0];
    const float* conv_w = (const float*)d_in[1];
    float* out = (float*)d_out;

    float* avgm = (float*)d_ws;          // NPOS_ floats (1 MiB)
    float* maxm = avgm + NPOS_;          // NPOS_ floats (1 MiB)
    float* ymap = maxm + NPOS_;          // NPOS_ floats (1 MiB)

    // 262144 positions / 16 per wave / 8 waves per block = 2048 blocks
    sa_reduce_kernel<<<NPOS_ / 128, 256, 0, stream>>>(x, avgm, maxm);
    // 262144 pixels / 256 = 1024 blocks
    sa_conv_kernel<<<NPOS_ / 256, 256, 0, stream>>>(avgm, maxm, conv_w, ymap);
    // 67108864 / 4 / 256 = 65536 blocks
    sa_scale_kernel<<<NTOT_ / 1024, 256, 0, stream>>>(x, ymap, out);
}